// TranslationBased_62027917688872
// MI455X (gfx1250) — compile-verified
//
#include <hip/hip_runtime.h>

typedef float v2f __attribute__((ext_vector_type(2)));
typedef float v8f __attribute__((ext_vector_type(8)));

#define LOG2E   1.4426950408889634f
#define M_MAX   512      // support points per feature (harness: 512)

// speed[n,d] = sum_f Cont[f] * sum_m exp(-||P[n]-S[f,m]||^2) * V[f,m,d]
//
// One wave32 owns a 16-row n-tile of Points (A matrix, resident in 2 VGPRs).
// Per f: block stages supp data into LDS as 32-byte records
//     rec[m] = { s0,s1,s2, y2 } , { Cont*v0, Cont*v1, Cont*v2, 0 }
// Inner loop per 16-col m-tile:
//     2 ds_load -> B tile via 2 cndmasks -> v_wmma_f32_16x16x4_f32 (K=4, f32)
//     8 args (distinct regs) -> 8 v_exp_f32 -> 12 v_pk_fma_f32 accumulates
// Epilogue: 16-lane butterfly per half-wave, lanes 0/16 store 8 rows x 3.
__global__ __launch_bounds__(256) void TranslationBased_speed_kernel(
    const float* __restrict__ Points,    // [N,3]
    const float* __restrict__ supp_pts,  // [F,M,3]
    const float* __restrict__ supp_vec,  // [F,M,3]
    const float* __restrict__ Cont,      // [F]
    float* __restrict__ out,             // [N,3]
    int N, int F, int M)
{
    // +16 records of padding so the tail prefetch of the double buffer
    // reads valid (ignored) LDS instead of out-of-range addresses.
    __shared__ float4 smem[(M_MAX + 16) * 2];   // 16.5 KB

    const int lane = threadIdx.x & 31;
    const int wave = threadIdx.x >> 5;
    const int tile = blockIdx.x * (blockDim.x >> 5) + wave;   // wave-uniform
    const int n_base = tile << 4;

    const int h   = lane >> 4;   // half-wave: K-pair {0,1} or {2,3}
    const int col = lane & 15;   // tile column (A: row n, B: col m)

    // ---- A matrix (Points tile), loaded once per wave ----
    int arow = n_base + col;
    if (arow >= N) arow = N - 1;          // lane-level clamp, loads only
    const float p0 = Points[arow * 3 + 0];
    const float p1 = Points[arow * 3 + 1];
    const float p2 = Points[arow * 3 + 2];
    v2f a;
    a.x = h ? p2 : p0;                    // K = 2h+0
    a.y = h ? 0.0f : p1;                  // K = 2h+1 (K=3 column = zero pad)
    const float x2 = p0 * p0 + p1 * p1 + p2 * p2;

    // x2p[v] = -log2e * x2 of row (n_base + 8h + v); rows live on lanes 0..15
    float x2p[8];
#pragma unroll
    for (int v = 0; v < 8; ++v)
        x2p[v] = -LOG2E * __shfl(x2, 8 * h + v, 32);

    float acc[8][3];
#pragma unroll
    for (int v = 0; v < 8; ++v) { acc[v][0] = 0.f; acc[v][1] = 0.f; acc[v][2] = 0.f; }

    const int mtiles = M >> 4;            // M/16 (M = 512 -> 32)

    for (int f = 0; f < F; ++f) {
        // ---------- stage supp data for this f into LDS ----------
        __syncthreads();                  // previous f fully consumed
        {
            const float cont = Cont[f];
            for (int idx = threadIdx.x; idx < M; idx += blockDim.x) {
                const float* sp = supp_pts + ((size_t)f * M + idx) * 3;
                const float* sv = supp_vec + ((size_t)f * M + idx) * 3;
                const float s0 = sp[0], s1 = sp[1], s2 = sp[2];
                float4 r0, r1;
                r0.x = s0; r0.y = s1; r0.z = s2;
                r0.w = s0 * s0 + s1 * s1 + s2 * s2;       // y2
                r1.x = cont * sv[0];
                r1.y = cont * sv[1];
                r1.z = cont * sv[2];
                r1.w = 0.0f;
                smem[idx * 2 + 0] = r0;
                smem[idx * 2 + 1] = r1;
            }
        }
        __syncthreads();

        // ---------- m-tile loop, double-buffered LDS reads ----------
        const float4* rec = smem + col * 2;     // this lane's m-column record
        float4 spq = rec[0];
        float4 wvq = rec[1];

        for (int mt = 0; mt < mtiles; ++mt) {
            rec += 32;                          // next tile: 16 records ahead
            const float4 spn = rec[0];          // prefetch (pad tile on last)
            const float4 wvn = rec[1];

            // ---- B matrix (supp_pts tile) ----
            v2f b;
            b.x = h ? spq.z : spq.x;
            b.y = h ? 0.0f : spq.y;

            // xy tile = A x B (+0):  v_wmma_f32_16x16x4_f32
            v8f d = {};
            d = __builtin_amdgcn_wmma_f32_16x16x4_f32(
                    /*neg_a=*/false, a, /*neg_b=*/false, b,
                    /*c_mod=*/(short)0, d, /*reuse_a=*/false, /*reuse_b=*/false);

            const float y2 = spq.w;
            const float w0 = wvq.x, w1 = wvq.y, w2 = wvq.z;

            // Phase 1: all 8 exp args in distinct registers (no WAR on the
            // trans source -> no forced v_nop after each v_exp_f32).
            float arg[8];
#pragma unroll
            for (int v = 0; v < 8; ++v)
                arg[v] = fmaf(d[v], 2.0f * LOG2E, fmaf(y2, -LOG2E, x2p[v]));

            // Phase 2: 8 independent transcendentals.
            float K[8];
#pragma unroll
            for (int v = 0; v < 8; ++v)
                K[v] = __builtin_amdgcn_exp2f(arg[v]);

            // Phase 3: accumulate, walked in v-pairs sharing a broadcast w
            // so the backend packs v_pk_fma_f32 without mov shuffling.
#pragma unroll
            for (int j = 0; j < 4; ++j) {
                const float K0 = K[2 * j], K1 = K[2 * j + 1];
                acc[2 * j][0]     = fmaf(K0, w0, acc[2 * j][0]);
                acc[2 * j + 1][0] = fmaf(K1, w0, acc[2 * j + 1][0]);
                acc[2 * j][1]     = fmaf(K0, w1, acc[2 * j][1]);
                acc[2 * j + 1][1] = fmaf(K1, w1, acc[2 * j + 1][1]);
                acc[2 * j][2]     = fmaf(K0, w2, acc[2 * j][2]);
                acc[2 * j + 1][2] = fmaf(K1, w2, acc[2 * j + 1][2]);
            }

            spq = spn;
            wvq = wvn;
        }
    }

    // ---- reduce over the 16 m-columns (xor masks < 16 keep halves separate) ----
#pragma unroll
    for (int mask = 1; mask < 16; mask <<= 1) {
#pragma unroll
        for (int v = 0; v < 8; ++v) {
            acc[v][0] += __shfl_xor(acc[v][0], mask, 32);
            acc[v][1] += __shfl_xor(acc[v][1], mask, 32);
            acc[v][2] += __shfl_xor(acc[v][2], mask, 32);
        }
    }

    // lane 0 writes rows n_base+0..7, lane 16 writes rows n_base+8..15
    if (col == 0) {
#pragma unroll
        for (int v = 0; v < 8; ++v) {
            const int n = n_base + 8 * h + v;
            if (n < N) {
                out[n * 3 + 0] = acc[v][0];
                out[n * 3 + 1] = acc[v][1];
                out[n * 3 + 2] = acc[v][2];
            }
        }
    }
}

extern "C" void kernel_launch(void* const* d_in, const int* in_sizes, int n_in,
                              void* d_out, int out_size, void* d_ws, size_t ws_size,
                              hipStream_t stream) {
    const float* Points   = (const float*)d_in[0];
    const float* supp_pts = (const float*)d_in[1];
    const float* supp_vec = (const float*)d_in[2];
    const float* Cont     = (const float*)d_in[3];
    float* out = (float*)d_out;

    const int N = in_sizes[0] / 3;               // 65536
    const int F = in_sizes[3];                   // 8
    const int M = in_sizes[1] / (3 * F);         // 512  (must be <= M_MAX)

    const int tiles = (N + 15) / 16;             // one wave per 16-row n-tile
    const int waves_per_block = 8;               // 256 threads = 8 wave32
    const int blocks = (tiles + waves_per_block - 1) / waves_per_block;

    TranslationBased_speed_kernel<<<blocks, waves_per_block * 32, 0, stream>>>(
        Points, supp_pts, supp_vec, Cont, out, N, F, M);
}